// GINE_47167330845239
// MI455X (gfx1250) — compile-verified
//
#include <hip/hip_runtime.h>
#include <math.h>

#define N_NODES 100000
#define N_EDGES 1600000
#define C       128
#define ECH     16
#define NLAYER  3
#define NGRAPH  256
#define GN_EPS  1e-5f

typedef __attribute__((ext_vector_type(2))) float v2f;
typedef __attribute__((ext_vector_type(8))) float v8f;

__device__ __forceinline__ float silu_f(float v) { return v / (1.0f + __expf(-v)); }

// ---------------------------------------------------------------- utilities
__global__ __launch_bounds__(256) void fill_zero_k(float* p, int n) {
    int i = blockIdx.x * 256 + threadIdx.x;
    if (i < n) p[i] = 0.0f;
}

__global__ __launch_bounds__(256) void count_nodes_k(const long long* __restrict__ batch,
                                                     float* __restrict__ cnt) {
    int i = blockIdx.x * 256 + threadIdx.x;
    if (i < N_NODES) unsafeAtomicAdd(&cnt[(int)batch[i]], 1.0f);
}

// ------------------------------------------------- edge message + scatter-add
// One wave per edge; lane owns 4 channels. e recomputed from edge_attr (16 ch)
// via LDS-resident 16x128 weight: 64 FMAs/lane. x[src] gather is L2-resident.
__global__ __launch_bounds__(256) void edge_aggr_k(
    const float* __restrict__ xin, const float* __restrict__ ea,
    const long long* __restrict__ eidx,
    const float* __restrict__ We, const float* __restrict__ be,
    float* __restrict__ aggr)
{
    __shared__ float sW[ECH * C];
    __shared__ float sb[C];
    for (int i = threadIdx.x; i < ECH * C; i += 256) sW[i] = We[i];
    for (int i = threadIdx.x; i < C; i += 256) sb[i] = be[i];
    __syncthreads();

    int lane = threadIdx.x & 31;
    int eid  = (blockIdx.x * 256 + threadIdx.x) >> 5;
    if (eid >= N_EDGES) return;

    int src = (int)eidx[eid];
    int dst = (int)eidx[(size_t)N_EDGES + eid];
    int c0  = lane * 4;

    float a0 = sb[c0], a1 = sb[c0 + 1], a2 = sb[c0 + 2], a3 = sb[c0 + 3];
    const float* earow = ea + (size_t)eid * ECH;
#pragma unroll
    for (int k = 0; k < ECH; ++k) {
        float av = earow[k];                       // wave-uniform, L0 broadcast
        const float4 wv = *(const float4*)&sW[k * C + c0];
        a0 += av * wv.x; a1 += av * wv.y; a2 += av * wv.z; a3 += av * wv.w;
    }
    const float4 xv = *(const float4*)(xin + (size_t)src * C + c0);
    float m0 = fmaxf(xv.x + silu_f(a0), 0.0f);
    float m1 = fmaxf(xv.y + silu_f(a1), 0.0f);
    float m2 = fmaxf(xv.z + silu_f(a2), 0.0f);
    float m3 = fmaxf(xv.w + silu_f(a3), 0.0f);

    float* ap = aggr + (size_t)dst * C + c0;
    unsafeAtomicAdd(ap + 0, m0);
    unsafeAtomicAdd(ap + 1, m1);
    unsafeAtomicAdd(ap + 2, m2);
    unsafeAtomicAdd(ap + 3, m3);
}

// ------------------------------------------------------- fused MLP via WMMA
// Block: 256 threads = 8 waves; processes 64 rows (4 tiles of 16).
// W1,W2 (64 KB each) staged in LDS once per block. Wave w owns output columns
// [16w,16w+16). h = x + aggr -> t = silu(h@W1+b1) -> y = t@W2+b2, all fp32
// through V_WMMA_F32_16X16X4_F32, bias folded into the accumulator.
#define HS (C + 4) /* 132: pad LDS row stride to kill bank conflicts */

__global__ __launch_bounds__(256) void mlp_wmma_k(
    const float* __restrict__ xin, const float* __restrict__ aggr,
    const float* __restrict__ W1, const float* __restrict__ b1,
    const float* __restrict__ W2, const float* __restrict__ b2,
    float* __restrict__ hout)
{
    __shared__ float sW1[C * C];
    __shared__ float sW2[C * C];
    __shared__ float sB1[C], sB2[C];
    __shared__ float sH[16 * HS];
    __shared__ float sT[16 * HS];

    for (int i = threadIdx.x * 4; i < C * C; i += 256 * 4) {
        *(float4*)&sW1[i] = *(const float4*)&W1[i];
        *(float4*)&sW2[i] = *(const float4*)&W2[i];
    }
    for (int i = threadIdx.x; i < C; i += 256) { sB1[i] = b1[i]; sB2[i] = b2[i]; }
    __syncthreads();

    const int wave = threadIdx.x >> 5;   // 0..7 -> 16-col slab
    const int lane = threadIdx.x & 31;
    const int n    = lane & 15;          // output column within slab / A row
    const int half = lane >> 4;          // K-pair select (ISA A-layout)
    const int c0   = wave * 16;
    const int rowBase = blockIdx.x * 64;

    for (int t = 0; t < 4; ++t) {
        int r0 = rowBase + t * 16;
        // stage h = x + aggr (zero-padded tail)
        for (int i = threadIdx.x; i < 16 * (C / 4); i += 256) {
            int rr = i >> 5;
            int cc = (i & 31) * 4;
            int row = r0 + rr;
            float4 v = make_float4(0.f, 0.f, 0.f, 0.f);
            if (row < N_NODES) {
                float4 xv = *(const float4*)(xin  + (size_t)row * C + cc);
                float4 av = *(const float4*)(aggr + (size_t)row * C + cc);
                v = make_float4(xv.x + av.x, xv.y + av.y, xv.z + av.z, xv.w + av.w);
            }
            *(float4*)&sH[rr * HS + cc] = v;
        }
        __syncthreads();

        // GEMM1 + SiLU
        v8f acc;
        {
            float bias = sB1[c0 + n];
#pragma unroll
            for (int j = 0; j < 8; ++j) acc[j] = bias;
        }
#pragma unroll
        for (int k0 = 0; k0 < C; k0 += 4) {
            int k = k0 + half * 2;
            v2f a = *(const v2f*)&sH[n * HS + k];
            v2f b;
            b.x = sW1[k * C + c0 + n];
            b.y = sW1[(k + 1) * C + c0 + n];
            acc = __builtin_amdgcn_wmma_f32_16x16x4_f32(
                false, a, false, b, (short)0, acc, false, false);
        }
#pragma unroll
        for (int j = 0; j < 8; ++j)
            sT[(j + half * 8) * HS + c0 + n] = silu_f(acc[j]);
        __syncthreads();

        // GEMM2
        {
            float bias = sB2[c0 + n];
#pragma unroll
            for (int j = 0; j < 8; ++j) acc[j] = bias;
        }
#pragma unroll
        for (int k0 = 0; k0 < C; k0 += 4) {
            int k = k0 + half * 2;
            v2f a = *(const v2f*)&sT[n * HS + k];
            v2f b;
            b.x = sW2[k * C + c0 + n];
            b.y = sW2[(k + 1) * C + c0 + n];
            acc = __builtin_amdgcn_wmma_f32_16x16x4_f32(
                false, a, false, b, (short)0, acc, false, false);
        }
#pragma unroll
        for (int j = 0; j < 8; ++j) {
            int row = r0 + j + half * 8;
            if (row < N_NODES) hout[(size_t)row * C + c0 + n] = acc[j];
        }
        __syncthreads();
    }
}

// ------------------------------------------------------ GraphNorm statistics
// batch is sorted: accumulate per-thread while graph id unchanged, flush one
// atomic per (graph-change, 4 channels).
__global__ __launch_bounds__(256) void gn_sum_k(
    const float* __restrict__ h, const long long* __restrict__ batch,
    float* __restrict__ S1)
{
    int c0 = (threadIdx.x & 31) * 4;
    int rg = threadIdx.x >> 5;
    int base = blockIdx.x * 64;
    float ax = 0, ay = 0, az = 0, aw = 0;
    int curg = -1;
    for (int i = rg; i < 64; i += 8) {
        int row = base + i;
        if (row >= N_NODES) break;
        int g = (int)batch[row];
        if (g != curg) {
            if (curg >= 0) {
                unsafeAtomicAdd(&S1[curg * C + c0 + 0], ax);
                unsafeAtomicAdd(&S1[curg * C + c0 + 1], ay);
                unsafeAtomicAdd(&S1[curg * C + c0 + 2], az);
                unsafeAtomicAdd(&S1[curg * C + c0 + 3], aw);
            }
            curg = g; ax = ay = az = aw = 0.f;
        }
        float4 v = *(const float4*)(h + (size_t)row * C + c0);
        ax += v.x; ay += v.y; az += v.z; aw += v.w;
    }
    if (curg >= 0) {
        unsafeAtomicAdd(&S1[curg * C + c0 + 0], ax);
        unsafeAtomicAdd(&S1[curg * C + c0 + 1], ay);
        unsafeAtomicAdd(&S1[curg * C + c0 + 2], az);
        unsafeAtomicAdd(&S1[curg * C + c0 + 3], aw);
    }
}

__global__ __launch_bounds__(256) void gn_sumsq_k(
    const float* __restrict__ h, const long long* __restrict__ batch,
    const float* __restrict__ S1, const float* __restrict__ cnt,
    const float* __restrict__ ms, float* __restrict__ S2)
{
    int c0 = (threadIdx.x & 31) * 4;
    int rg = threadIdx.x >> 5;
    int base = blockIdx.x * 64;
    float ax = 0, ay = 0, az = 0, aw = 0;
    float mx = 0, my = 0, mz = 0, mw = 0;
    int curg = -1;
    for (int i = rg; i < 64; i += 8) {
        int row = base + i;
        if (row >= N_NODES) break;
        int g = (int)batch[row];
        if (g != curg) {
            if (curg >= 0) {
                unsafeAtomicAdd(&S2[curg * C + c0 + 0], ax);
                unsafeAtomicAdd(&S2[curg * C + c0 + 1], ay);
                unsafeAtomicAdd(&S2[curg * C + c0 + 2], az);
                unsafeAtomicAdd(&S2[curg * C + c0 + 3], aw);
            }
            curg = g; ax = ay = az = aw = 0.f;
            float ic = 1.0f / cnt[g];
            mx = S1[g * C + c0 + 0] * ic * ms[c0 + 0];
            my = S1[g * C + c0 + 1] * ic * ms[c0 + 1];
            mz = S1[g * C + c0 + 2] * ic * ms[c0 + 2];
            mw = S1[g * C + c0 + 3] * ic * ms[c0 + 3];
        }
        float4 v = *(const float4*)(h + (size_t)row * C + c0);
        float dx = v.x - mx, dy = v.y - my, dz = v.z - mz, dw = v.w - mw;
        ax += dx * dx; ay += dy * dy; az += dz * dz; aw += dw * dw;
    }
    if (curg >= 0) {
        unsafeAtomicAdd(&S2[curg * C + c0 + 0], ax);
        unsafeAtomicAdd(&S2[curg * C + c0 + 1], ay);
        unsafeAtomicAdd(&S2[curg * C + c0 + 2], az);
        unsafeAtomicAdd(&S2[curg * C + c0 + 3], aw);
    }
}

__global__ __launch_bounds__(256) void gn_apply_k(
    const float* __restrict__ h, const long long* __restrict__ batch,
    const float* __restrict__ S1, const float* __restrict__ S2,
    const float* __restrict__ cnt,
    const float* __restrict__ w, const float* __restrict__ bgn,
    const float* __restrict__ ms, float* __restrict__ xout)
{
    int idx = blockIdx.x * 256 + threadIdx.x;
    int row = idx >> 5;
    if (row >= N_NODES) return;
    int c0 = (idx & 31) * 4;
    int g = (int)batch[row];
    float ic = 1.0f / cnt[g];
    float4 v = *(const float4*)(h + (size_t)row * C + c0);
    float4 o;
    {
        float mean = S1[g * C + c0 + 0] * ic;
        float var  = S2[g * C + c0 + 0] * ic;
        o.x = w[c0 + 0] * (v.x - mean * ms[c0 + 0]) * rsqrtf(var + GN_EPS) + bgn[c0 + 0];
    }
    {
        float mean = S1[g * C + c0 + 1] * ic;
        float var  = S2[g * C + c0 + 1] * ic;
        o.y = w[c0 + 1] * (v.y - mean * ms[c0 + 1]) * rsqrtf(var + GN_EPS) + bgn[c0 + 1];
    }
    {
        float mean = S1[g * C + c0 + 2] * ic;
        float var  = S2[g * C + c0 + 2] * ic;
        o.z = w[c0 + 2] * (v.z - mean * ms[c0 + 2]) * rsqrtf(var + GN_EPS) + bgn[c0 + 2];
    }
    {
        float mean = S1[g * C + c0 + 3] * ic;
        float var  = S2[g * C + c0 + 3] * ic;
        o.w = w[c0 + 3] * (v.w - mean * ms[c0 + 3]) * rsqrtf(var + GN_EPS) + bgn[c0 + 3];
    }
    *(float4*)(xout + (size_t)row * C + c0) = o;
}

// ---------------------------------------------------------------- launcher
extern "C" void kernel_launch(void* const* d_in, const int* in_sizes, int n_in,
                              void* d_out, int out_size, void* d_ws, size_t ws_size,
                              hipStream_t stream) {
    const float*     x_in  = (const float*)d_in[0];
    const float*     ea    = (const float*)d_in[1];
    const long long* eidx  = (const long long*)d_in[2];
    const long long* batch = (const long long*)d_in[3];
    const float*     lew   = (const float*)d_in[4];
    const float*     leb   = (const float*)d_in[5];
    const float*     cw1   = (const float*)d_in[6];
    const float*     cb1   = (const float*)d_in[7];
    const float*     cw2   = (const float*)d_in[8];
    const float*     cb2   = (const float*)d_in[9];
    const float*     gnw   = (const float*)d_in[10];
    const float*     gnb   = (const float*)d_in[11];
    const float*     gnms  = (const float*)d_in[12];

    const size_t nc = (size_t)N_NODES * C;
    float* aggr = (float*)d_ws;
    float* hbuf = aggr + nc;
    float* xbuf = hbuf + nc;
    float* S1   = xbuf + nc;
    float* S2   = S1 + NGRAPH * C;
    float* cnt  = S2 + NGRAPH * C;

    // per-graph node counts (once per call)
    fill_zero_k<<<1, 256, 0, stream>>>(cnt, NGRAPH);
    count_nodes_k<<<(N_NODES + 255) / 256, 256, 0, stream>>>(batch, cnt);

    const int fillA  = (int)((nc + 255) / 256);
    const int edgeB  = (N_EDGES * 32 + 255) / 256;
    const int mlpB   = (N_NODES + 63) / 64;
    const int gnB    = (N_NODES + 63) / 64;
    const int applyB = (N_NODES * 32 + 255) / 256;

    for (int l = 0; l < NLAYER; ++l) {
        const float* xsrc = (l == 0) ? x_in : xbuf;
        float*       xdst = (l == NLAYER - 1) ? (float*)d_out : xbuf;

        fill_zero_k<<<fillA, 256, 0, stream>>>(aggr, (int)nc);
        fill_zero_k<<<(2 * NGRAPH * C + 255) / 256, 256, 0, stream>>>(S1, 2 * NGRAPH * C);

        edge_aggr_k<<<edgeB, 256, 0, stream>>>(xsrc, ea, eidx, lew, leb, aggr);

        mlp_wmma_k<<<mlpB, 256, 0, stream>>>(
            xsrc, aggr,
            cw1 + (size_t)l * C * C, cb1 + (size_t)l * C,
            cw2 + (size_t)l * C * C, cb2 + (size_t)l * C,
            hbuf);

        gn_sum_k  <<<gnB, 256, 0, stream>>>(hbuf, batch, S1);
        gn_sumsq_k<<<gnB, 256, 0, stream>>>(hbuf, batch, S1, cnt, gnms + (size_t)l * C, S2);
        gn_apply_k<<<applyB, 256, 0, stream>>>(hbuf, batch, S1, S2, cnt,
                                               gnw + (size_t)l * C, gnb + (size_t)l * C,
                                               gnms + (size_t)l * C, xdst);
    }
}